// QuantumLayer_1864015807130
// MI455X (gfx1250) — compile-verified
//
#include <hip/hip_runtime.h>

// ---------------------------------------------------------------------------
// QuantumLayer on MI455X (gfx1250, wave32, WMMA).
//
// Math: psi_final = U' @ r  where U' is the full 6-layer circuit unitary with
// the AngleEmbedding phases (-i)^popcount folded into its columns, and r is
// the real magnitude vector r[s] = prod_q (bit ? sin(x_q/2) : cos(x_q/2)).
// EV_q = sum_s sign_q(s) * (psi_r[s]^2 + psi_i[s]^2).
//
// WMMA mapping (V_WMMA_F32_16X16X4_F32, D = A x B + C):
//   A (16x4 f32)  = K-chunk of Re(U') / Im(U'); lane L (M=L&15):
//                   VGPR v holds K = 4*chunk + v + (L>=16 ? 2 : 0)
//   B (4x16 f32)  = r columns; lane L (N=L&15):
//                   VGPR v holds K = 4*chunk + v + (L>=16 ? 2 : 0)
//   D (16x16 f32) = psi^T; lane L, VGPR rr -> (M = rr + (L>=16?8:0), N = L&15)
// So each sample's 16 amplitudes live in lanes n and n+16: in-lane butterfly
// + one __shfl_xor(16) gives all four Pauli-Z expectation values.
// ---------------------------------------------------------------------------

typedef __attribute__((ext_vector_type(2))) float v2f;
typedef __attribute__((ext_vector_type(8))) float v8f;

#define N_LAYERS 6

// ---------------------------------------------------------------------------
// Kernel 1: build U' from weights (6,4,3) and emit it pre-swizzled into the
// WMMA A-operand layout: Apacked[lane*16 + idx],
// idx = {part(Re/Im)}<<3 | chunk<<1 | vgpr.
// ---------------------------------------------------------------------------
__global__ __launch_bounds__(256) void build_unitary_kernel(
    const float* __restrict__ w, float* __restrict__ Apacked) {
  __shared__ float Ur[256];
  __shared__ float Ui[256];
  const int t = threadIdx.x;       // 256 threads: one per U entry
  const int row = t >> 4;          // output state index (0..15)
  const int col = t & 15;          // input state index  (0..15)

  Ur[t] = (row == col) ? 1.0f : 0.0f;
  Ui[t] = 0.0f;
  __syncthreads();

  const int cq[4] = {0, 1, 2, 3};  // CNOT control qubits
  const int tq[4] = {1, 2, 3, 0};  // CNOT target qubits

  for (int l = 0; l < N_LAYERS; ++l) {
    // CNOT ring (qubit q <-> state bit 3-q)
    for (int p = 0; p < 4; ++p) {
      const int cm = 1 << (3 - cq[p]);
      const int tm = 1 << (3 - tq[p]);
      const int src = (row & cm) ? (row ^ tm) : row;
      const float vr = Ur[src * 16 + col];
      const float vi = Ui[src * 16 + col];
      __syncthreads();
      Ur[t] = vr; Ui[t] = vi;
      __syncthreads();
    }
    // Per-qubit RX, RY, RZ (reference order)
    for (int q = 0; q < 4; ++q) {
      const int m  = 1 << (3 - q);
      const int b  = (row & m) ? 1 : 0;
      const int r0 = (row & ~m) * 16 + col;
      const int r1 = (row | m) * 16 + col;
      for (int g = 0; g < 3; ++g) {
        const float th = 0.5f * w[(l * 4 + q) * 3 + g];
        const float c = __cosf(th), s = __sinf(th);
        float g0r, g0i, g1r, g1i;  // gate row 'b': coeffs for old(bit=0), old(bit=1)
        if (g == 0) {        // RX: [[c,-is],[-is,c]]
          if (b == 0) { g0r = c; g0i = 0.f; g1r = 0.f; g1i = -s; }
          else        { g0r = 0.f; g0i = -s; g1r = c; g1i = 0.f; }
        } else if (g == 1) { // RY: [[c,-s],[s,c]]
          if (b == 0) { g0r = c;  g0i = 0.f; g1r = -s; g1i = 0.f; }
          else        { g0r = s;  g0i = 0.f; g1r = c;  g1i = 0.f; }
        } else {             // RZ: diag(c-is, c+is)
          if (b == 0) { g0r = c;  g0i = -s;  g1r = 0.f; g1i = 0.f; }
          else        { g0r = 0.f; g0i = 0.f; g1r = c;  g1i = s;  }
        }
        const float ar = Ur[r0], ai = Ui[r0];
        const float br = Ur[r1], bi = Ui[r1];
        const float nr = g0r * ar - g0i * ai + g1r * br - g1i * bi;
        const float ni = g0r * ai + g0i * ar + g1r * bi + g1i * br;
        __syncthreads();
        Ur[t] = nr; Ui[t] = ni;
        __syncthreads();
      }
    }
  }

  // Fold AngleEmbedding phases into columns: U'[:,j] = U[:,j] * (-i)^popcount(j)
  {
    const int pc = __popc(col) & 3;
    float pr, pi;
    if      (pc == 0) { pr =  1.f; pi =  0.f; }
    else if (pc == 1) { pr =  0.f; pi = -1.f; }
    else if (pc == 2) { pr = -1.f; pi =  0.f; }
    else              { pr =  0.f; pi =  1.f; }
    const float ur = Ur[t], ui = Ui[t];
    const float nr = ur * pr - ui * pi;
    const float ni = ur * pi + ui * pr;
    __syncthreads();
    Ur[t] = nr; Ui[t] = ni;
    __syncthreads();
  }

  // Pack into WMMA A-operand lane order (512 floats, 2 per thread)
  for (int e = 2 * t; e < 2 * t + 2; ++e) {
    const int L    = e >> 4;
    const int idx  = e & 15;
    const int part = idx >> 3;        // 0 = Re, 1 = Im
    const int c    = (idx >> 1) & 3;  // K chunk
    const int v    = idx & 1;         // VGPR within chunk
    const int M    = L & 15;
    const int K    = 4 * c + v + ((L >= 16) ? 2 : 0);
    Apacked[e] = part ? Ui[M * 16 + K] : Ur[M * 16 + K];
  }
}

// ---------------------------------------------------------------------------
// Kernel 2: one wave per 16-sample tile; 8 x v_wmma_f32_16x16x4_f32 per tile.
// ---------------------------------------------------------------------------
__global__ __launch_bounds__(256) void qnn_wmma_kernel(
    const float* __restrict__ x, const float* __restrict__ Apacked,
    float* __restrict__ out, int nTiles) {
  const int lane = threadIdx.x & 31;
  const int wavesPerBlock = blockDim.x >> 5;
  const int wave   = blockIdx.x * wavesPerBlock + (threadIdx.x >> 5);
  const int nWaves = gridDim.x * wavesPerBlock;

  // Constant A registers for this kernel (hot in L0/L2: 2 KB total)
  v2f aRe[4], aIm[4];
  {
    const float4* ap = (const float4*)(Apacked + (size_t)lane * 16);
    const float4 f0 = ap[0], f1 = ap[1], f2 = ap[2], f3 = ap[3];
    aRe[0] = v2f{f0.x, f0.y}; aRe[1] = v2f{f0.z, f0.w};
    aRe[2] = v2f{f1.x, f1.y}; aRe[3] = v2f{f1.z, f1.w};
    aIm[0] = v2f{f2.x, f2.y}; aIm[1] = v2f{f2.z, f2.w};
    aIm[2] = v2f{f3.x, f3.y}; aIm[3] = v2f{f3.z, f3.w};
  }

  const int  n  = lane & 15;
  const bool hi = lane >= 16;

  for (int tile = wave; tile < nTiles; tile += nWaves) {
    const int bidx = tile * 16 + n;  // this lane's sample (duplicated in lane n+16)
    const float4 xv = *(const float4*)(x + (size_t)bidx * 4);

    const float c0 = __cosf(0.5f * xv.x), s0 = __sinf(0.5f * xv.x);
    const float c1 = __cosf(0.5f * xv.y), s1 = __sinf(0.5f * xv.y);
    const float c2 = __cosf(0.5f * xv.z), s2 = __sinf(0.5f * xv.z);
    const float c3 = __cosf(0.5f * xv.w), s3 = __sinf(0.5f * xv.w);

    // r[s] = t[s>>2] * u[s&3]; this lane only needs u[v + (hi?2:0)]
    const float t0 = c0 * c1, t1 = c0 * s1, t2 = s0 * c1, t3 = s0 * s1;
    const float q2 = hi ? s2 : c2;
    const float ua = q2 * c3;   // u[hi?2:0]
    const float ub = q2 * s3;   // u[hi?3:1]

    const v2f b0 = v2f{t0 * ua, t0 * ub};
    const v2f b1 = v2f{t1 * ua, t1 * ub};
    const v2f b2 = v2f{t2 * ua, t2 * ub};
    const v2f b3 = v2f{t3 * ua, t3 * ub};

    v8f accR = {}; v8f accI = {};
    // Interleave real/imag chains so consecutive WMMAs are independent.
    accR = __builtin_amdgcn_wmma_f32_16x16x4_f32(false, aRe[0], false, b0, (short)0, accR, false, false);
    accI = __builtin_amdgcn_wmma_f32_16x16x4_f32(false, aIm[0], false, b0, (short)0, accI, false, false);
    accR = __builtin_amdgcn_wmma_f32_16x16x4_f32(false, aRe[1], false, b1, (short)0, accR, false, false);
    accI = __builtin_amdgcn_wmma_f32_16x16x4_f32(false, aIm[1], false, b1, (short)0, accI, false, false);
    accR = __builtin_amdgcn_wmma_f32_16x16x4_f32(false, aRe[2], false, b2, (short)0, accR, false, false);
    accI = __builtin_amdgcn_wmma_f32_16x16x4_f32(false, aIm[2], false, b2, (short)0, accI, false, false);
    accR = __builtin_amdgcn_wmma_f32_16x16x4_f32(false, aRe[3], false, b3, (short)0, accR, false, false);
    accI = __builtin_amdgcn_wmma_f32_16x16x4_f32(false, aIm[3], false, b3, (short)0, accI, false, false);

    // Probabilities for in-lane states s = rr + (hi?8:0)
    float p[8];
#pragma unroll
    for (int rr = 0; rr < 8; ++rr)
      p[rr] = accR[rr] * accR[rr] + accI[rr] * accI[rr];

    // Walsh butterfly over rr (state bits 2..0 = qubits 1..3)
    const float u0 = p[0] + p[1], w0 = p[0] - p[1];
    const float u1 = p[2] + p[3], w1 = p[2] - p[3];
    const float u2 = p[4] + p[5], w2 = p[4] - p[5];
    const float u3 = p[6] + p[7], w3 = p[6] - p[7];
    const float T  = (u0 + u1) + (u2 + u3);  // plain sum         (for qubit0 via halves)
    const float S2 = (u0 + u1) - (u2 + u3);  // sign by rr&4      -> qubit1
    const float S1 = (u0 - u1) + (u2 - u3);  // sign by rr&2      -> qubit2
    const float S0 = (w0 + w1) + (w2 + w3);  // sign by rr&1      -> qubit3

    // Merge state bit3 (qubit0) halves: lane n <-> lane n+16
    const float To  = __shfl_xor(T, 16, 32);
    const float S2o = __shfl_xor(S2, 16, 32);
    const float S1o = __shfl_xor(S1, 16, 32);
    const float S0o = __shfl_xor(S0, 16, 32);

    if (!hi) {
      float4 ev;
      ev.x = T - To;     // qubit0: +states 0..7, -states 8..15
      ev.y = S2 + S2o;   // qubit1
      ev.z = S1 + S1o;   // qubit2
      ev.w = S0 + S0o;   // qubit3
      *(float4*)(out + (size_t)bidx * 4) = ev;
    }
  }
}

// ---------------------------------------------------------------------------
extern "C" void kernel_launch(void* const* d_in, const int* in_sizes, int n_in,
                              void* d_out, int out_size, void* d_ws, size_t ws_size,
                              hipStream_t stream) {
  const float* x = (const float*)d_in[0];       // [BATCH, 4]
  const float* w = (const float*)d_in[1];       // [6, 4, 3]
  float* out     = (float*)d_out;               // [BATCH, 4]
  float* Apacked = (float*)d_ws;                // 512 floats (2 KB)

  build_unitary_kernel<<<1, 256, 0, stream>>>(w, Apacked);

  const int batch  = in_sizes[0] / 4;
  const int nTiles = batch / 16;                // 32768 for BATCH=524288
  const int threads = 256;                      // 8 waves/block
  const int wavesPerBlock = threads / 32;
  int blocks = (nTiles + 2 * wavesPerBlock - 1) / (2 * wavesPerBlock); // ~2 tiles/wave
  if (blocks < 1) blocks = 1;
  qnn_wmma_kernel<<<blocks, threads, 0, stream>>>(x, Apacked, out, nTiles);
}